// PortHamiltonianLayer_26010321945364
// MI455X (gfx1250) — compile-verified
//
#include <hip/hip_runtime.h>
#include <math.h>

typedef __attribute__((ext_vector_type(2))) float v2f;
typedef __attribute__((ext_vector_type(8))) float v8f;

#define DIM 128
#define HID 512
#define NTOK 2048
#define NTILE 128            // token tiles of 16 tokens
#define KBLOCKS 4            // split of k (0..127) across workgroups
#define KPERBLOCK 32
#define CHUNK_K 4            // k values per LDS chunk
#define CHUNK_ROWS (CHUNK_K * DIM)   // 512 rows of Wj/Wr per chunk
#define NCHUNK (KPERBLOCK / CHUNK_K) // 8
#define XIN_STRIDE 132       // padded row stride (floats) for 16xD LDS arrays
#define TILE_STRIDE 17       // padded token stride (floats) for the chunk tile
#define LN_EPS 1e-5f
#define R_EPS 1e-4f

// ---------------------------------------------------------------------------
// Kernel 1: LayerNorm + analytic potential-net gradient per 16-token tile.
//   x_in = LN(x; gamma, beta)
//   z = x_in W1^T + b1 ; gmul = silu'(z) * w2 ; grads = gmul @ W1
// ---------------------------------------------------------------------------
__global__ __launch_bounds__(256) void k_ln_grads(
    const float* __restrict__ x, const float* __restrict__ W1,
    const float* __restrict__ b1, const float* __restrict__ w2,
    const float* __restrict__ gamma, const float* __restrict__ beta,
    float* __restrict__ xin_ws, float* __restrict__ grads_ws)
{
  __shared__ float xin[16 * XIN_STRIDE];
  __shared__ float gmul[HID * TILE_STRIDE];   // [h][t], padded

  const int tid  = threadIdx.x;
  const int tile = blockIdx.x;

  // ---- LayerNorm: 16 threads per token, 8 elements each
  {
    const int tl = tid >> 4;           // token in tile
    const int j  = tid & 15;
    const int tg = tile * 16 + tl;
    const float* xp = x + (size_t)tg * DIM + j * 8;
    float v[8];
    float s = 0.f;
    #pragma unroll
    for (int e = 0; e < 8; ++e) { v[e] = xp[e]; s += v[e]; }
    #pragma unroll
    for (int m = 1; m < 16; m <<= 1) s += __shfl_xor(s, m, 32);
    const float mu = s * (1.0f / 128.0f);
    float s2 = 0.f;
    #pragma unroll
    for (int e = 0; e < 8; ++e) { float c = v[e] - mu; s2 += c * c; }
    #pragma unroll
    for (int m = 1; m < 16; m <<= 1) s2 += __shfl_xor(s2, m, 32);
    const float rstd = rsqrtf(s2 * (1.0f / 128.0f) + LN_EPS);
    #pragma unroll
    for (int e = 0; e < 8; ++e) {
      const int d = j * 8 + e;
      const float xn = (v[e] - mu) * rstd * gamma[d] + beta[d];
      xin[tl * XIN_STRIDE + d] = xn;
      xin_ws[(size_t)tg * DIM + d] = xn;
    }
  }
  __syncthreads();

  // ---- z / gmul: thread -> (h = tid>>4 + 16*it, t = tid&15)
  const int t  = tid & 15;
  const int g0 = tid >> 4;
  for (int it = 0; it < 32; ++it) {
    const int h = g0 + 16 * it;
    const float* wr = W1 + (size_t)h * DIM;
    const float* xr = xin + t * XIN_STRIDE;
    float z = b1[h];
    for (int d = 0; d < DIM; ++d) z += wr[d] * xr[d];
    const float sig = 1.0f / (1.0f + __expf(-z));
    const float dsl = sig * (1.0f + z * (1.0f - sig));
    gmul[h * TILE_STRIDE + t] = dsl * w2[h];
  }
  __syncthreads();

  // ---- grads[t][d] = sum_h gmul[h][t] * W1[h][d]
  for (int it = 0; it < 8; ++it) {
    const int d = g0 + 16 * it;
    float sacc = 0.f;
    for (int h = 0; h < HID; ++h)
      sacc += gmul[h * TILE_STRIDE + t] * W1[(size_t)h * DIM + d];
    grads_ws[(size_t)(tile * 16 + t) * DIM + d] = sacc;
  }
}

// ---------------------------------------------------------------------------
// WMMA weight pass: compute one chunk (CHUNK_ROWS x 16 tokens) of
// W[rows] @ x_in^T via V_WMMA_F32_16X16X4_F32, add bias, store into LDS tile.
// Layout assumptions (CDNA5 ISA 7.12.2, 32-bit A/B/C):
//   A 16x4: lanes 0-15 hold K=0,1 ; lanes 16-31 hold K=2,3
//   B 4x16: vgpr j -> K=j (lanes 0-15) / K=j+2 (lanes 16-31), N = lane&15
//   D 16x16: vgpr r -> M=r (lanes 0-15) / M=r+8 (lanes 16-31), N = lane&15
// ---------------------------------------------------------------------------
__device__ __forceinline__ void weight_pass(
    const float* __restrict__ Wt, const float* __restrict__ bt,
    int rows_base, const v2f* __restrict__ bfrag, float* __restrict__ tile)
{
  const int wid  = threadIdx.x >> 5;
  const int lane = threadIdx.x & 31;
  const int lt   = lane & 15;
  const int hi   = lane >> 4;
  const int MT_PER_WAVE = (CHUNK_ROWS / 16) / 8;   // 4

  #pragma unroll
  for (int m = 0; m < MT_PER_WAVE; ++m) {
    const int mt    = wid * MT_PER_WAVE + m;       // 0..31
    const int row_l = mt * 16 + lt;
    const float* arow = Wt + (size_t)(rows_base + row_l) * DIM + 2 * hi;
    v8f acc = {};
    #pragma unroll
    for (int ks = 0; ks < 32; ++ks) {
      v2f a = *(const v2f*)(arow + 4 * ks);
      acc = __builtin_amdgcn_wmma_f32_16x16x4_f32(
          false, a, false, bfrag[ks], (short)0, acc, false, false);
    }
    const int rb = mt * 16 + 8 * hi;
    #pragma unroll
    for (int r = 0; r < 8; ++r) {
      const float bias = bt[rows_base + rb + r];
      tile[(rb + r) * TILE_STRIDE + lt] = acc[r] + bias;
    }
  }
}

// ---------------------------------------------------------------------------
// Kernel 2: heavy fused pass.  update = M g - M^T g - P^T (P g)  (partial in k)
// One workgroup per (token tile, k-block of 32 k's); 8 chunks of 4 k's.
// ---------------------------------------------------------------------------
__global__ __launch_bounds__(256) void k_heavy(
    const float* __restrict__ Wj, const float* __restrict__ bj,
    const float* __restrict__ Wr, const float* __restrict__ br,
    const float* __restrict__ xin_ws, const float* __restrict__ grads_ws,
    float* __restrict__ part_ws)
{
  __shared__ float xin [16 * XIN_STRIDE];
  __shared__ float gr  [16 * XIN_STRIDE];
  __shared__ float uacc[16 * XIN_STRIDE];
  __shared__ float tile[CHUNK_ROWS * TILE_STRIDE];
  __shared__ float vsc [16 * CHUNK_K];

  const int tid   = threadIdx.x;
  const int kb    = blockIdx.x & (KBLOCKS - 1);
  const int ttile = blockIdx.x >> 2;

  for (int idx = tid; idx < 16 * DIM; idx += 256) {
    const int t = idx >> 7, d = idx & 127;
    xin [t * XIN_STRIDE + d] = xin_ws  [(size_t)(ttile * 16 + t) * DIM + d];
    gr  [t * XIN_STRIDE + d] = grads_ws[(size_t)(ttile * 16 + t) * DIM + d];
    uacc[t * XIN_STRIDE + d] = 0.f;
  }
  __syncthreads();

  // B fragments for the whole workgroup: x_in^T, 32 K-steps, kept in VGPRs.
  const int lane = tid & 31;
  const int lt = lane & 15, hi = lane >> 4;
  v2f bfrag[32];
  #pragma unroll
  for (int ks = 0; ks < 32; ++ks)
    bfrag[ks] = *(const v2f*)(&xin[lt * XIN_STRIDE + 4 * ks + 2 * hi]);

  for (int c = 0; c < NCHUNK; ++c) {
    const int kg0 = kb * KPERBLOCK + c * CHUNK_K;   // global k base of chunk
    const int rows_base = kg0 * DIM;

    // ---- M chunk (Wj) ----
    weight_pass(Wj, bj, rows_base, bfrag, tile);
    __syncthreads();

    // u1[t][k] = sum_i M[k,i] g[i]  -> uacc[t][kg] += u1
    if (tid < 16 * CHUNK_K) {
      const int t = tid >> 2, kl = tid & 3;
      float s = 0.f;
      for (int i = 0; i < DIM; ++i)
        s += tile[(kl * DIM + i) * TILE_STRIDE + t] * gr[t * XIN_STRIDE + i];
      uacc[t * XIN_STRIDE + kg0 + kl] += s;
    }
    __syncthreads();

    // u2[t][i] = sum_k M[k,i] g[k]  -> uacc[t][i] -= u2 (partial over chunk)
    {
      const int i = tid & 127, th = tid >> 7;
      for (int tt2 = 0; tt2 < 8; ++tt2) {
        const int t = th * 8 + tt2;
        float s = 0.f;
        #pragma unroll
        for (int kl = 0; kl < CHUNK_K; ++kl)
          s += tile[(kl * DIM + i) * TILE_STRIDE + t] * gr[t * XIN_STRIDE + kg0 + kl];
        uacc[t * XIN_STRIDE + i] -= s;
      }
    }
    __syncthreads();

    // ---- P chunk (Wr) ----
    weight_pass(Wr, br, rows_base, bfrag, tile);
    __syncthreads();

    // v[t][k] = sum_i P[k,i] g[i]
    if (tid < 16 * CHUNK_K) {
      const int t = tid >> 2, kl = tid & 3;
      float s = 0.f;
      for (int i = 0; i < DIM; ++i)
        s += tile[(kl * DIM + i) * TILE_STRIDE + t] * gr[t * XIN_STRIDE + i];
      vsc[t * CHUNK_K + kl] = s;
    }
    __syncthreads();

    // u3[t][i] = sum_k P[k,i] v[k]  -> uacc[t][i] -= u3 (partial over chunk)
    {
      const int i = tid & 127, th = tid >> 7;
      for (int tt2 = 0; tt2 < 8; ++tt2) {
        const int t = th * 8 + tt2;
        float s = 0.f;
        #pragma unroll
        for (int kl = 0; kl < CHUNK_K; ++kl)
          s += tile[(kl * DIM + i) * TILE_STRIDE + t] * vsc[t * CHUNK_K + kl];
        uacc[t * XIN_STRIDE + i] -= s;
      }
    }
    __syncthreads();
  }

  for (int idx = tid; idx < 16 * DIM; idx += 256) {
    const int t = idx >> 7, d = idx & 127;
    part_ws[(((size_t)kb * NTILE + ttile) * 16 + t) * DIM + d] =
        uacc[t * XIN_STRIDE + d];
  }
}

// ---------------------------------------------------------------------------
// Kernel 3: out = x - R_EPS * grads + sum_kb partial[kb]
// ---------------------------------------------------------------------------
__global__ __launch_bounds__(256) void k_combine(
    const float* __restrict__ x, const float* __restrict__ grads_ws,
    const float* __restrict__ part_ws, float* __restrict__ out)
{
  const int idx = blockIdx.x * 256 + threadIdx.x;   // 0 .. NTOK*DIM-1
  const int tg = idx >> 7, d = idx & 127;
  const int ttile = tg >> 4, t = tg & 15;
  float s = x[idx] - R_EPS * grads_ws[idx];
  #pragma unroll
  for (int kb = 0; kb < KBLOCKS; ++kb)
    s += part_ws[(((size_t)kb * NTILE + ttile) * 16 + t) * DIM + d];
  out[idx] = s;
}

extern "C" void kernel_launch(void* const* d_in, const int* in_sizes, int n_in,
                              void* d_out, int out_size, void* d_ws, size_t ws_size,
                              hipStream_t stream) {
  const float* x     = (const float*)d_in[0];
  const float* W1    = (const float*)d_in[1];
  const float* b1    = (const float*)d_in[2];
  const float* w2    = (const float*)d_in[3];
  // d_in[4] = b2 : no effect on the gradient, unused
  const float* Wj    = (const float*)d_in[5];
  const float* bj    = (const float*)d_in[6];
  const float* Wr    = (const float*)d_in[7];
  const float* br    = (const float*)d_in[8];
  const float* gamma = (const float*)d_in[9];
  const float* beta  = (const float*)d_in[10];
  float* out = (float*)d_out;

  float* wsf      = (float*)d_ws;
  float* xin_ws   = wsf;                         // NTOK*DIM
  float* grads_ws = wsf + (size_t)NTOK * DIM;    // NTOK*DIM
  float* part_ws  = wsf + (size_t)2 * NTOK * DIM;// KBLOCKS*NTOK*DIM

  k_ln_grads<<<NTILE, 256, 0, stream>>>(x, W1, b1, w2, gamma, beta,
                                        xin_ws, grads_ws);
  k_heavy<<<NTILE * KBLOCKS, 256, 0, stream>>>(Wj, bj, Wr, br,
                                               xin_ws, grads_ws, part_ws);
  k_combine<<<(NTOK * DIM) / 256, 256, 0, stream>>>(x, grads_ws, part_ws, out);
}